// SheafConvolutionalNetwork_81698867905240
// MI455X (gfx1250) — compile-verified
//
#include <hip/hip_runtime.h>
#include <hip/hip_bf16.h>
#include <math.h>

// ---------------------------------------------------------------------------
// CDNA5 (gfx1250) wave32 WMMA helpers: D = A(16x4,f32) x B(4x16,f32) + C
// A layout: lanes 0-15 -> (M=lane, K=0/1 in v0/v1), lanes 16-31 -> K=2/3
// B layout: v0 = row K=0 (lanes 0-15) / K=2 (lanes 16-31); v1 = K=1 / K=3
// C layout: vgpr v, lanes 0-15 -> (M=v, N=lane); lanes 16-31 -> (M=v+8, N=lane-16)
// ---------------------------------------------------------------------------
typedef float v2f __attribute__((ext_vector_type(2)));
typedef float v8f __attribute__((ext_vector_type(8)));

#define DEVINL __device__ __forceinline__

DEVINL v8f wmma4_f32(v2f a, v2f b, v8f c) {
    // 8-arg form: (neg_a, A, neg_b, B, c_mod, C, reuse_a, reuse_b)
    return __builtin_amdgcn_wmma_f32_16x16x4_f32(
        false, a, false, b, (short)0, c, false, false);
}

// ---------------------------------------------------------------------------
// Kernel 1: fused input embedding + G = relu(E@We+be) @ W2
//   E: (4096,128)  We: (128,64)  be: (64)  W2: (64,64)  ->  G: (4096,64)
// Block = 128 threads (4 waves), each block owns a 16-row tile; wave w owns
// output columns [16w,16w+16). Stage 1 WMMA writes relu(H) tile to LDS,
// stage 2 WMMA reads A-fragments from LDS (K=64).
// ---------------------------------------------------------------------------
__global__ void k_embed(const float* __restrict__ E, const float* __restrict__ We,
                        const float* __restrict__ be, const float* __restrict__ W2,
                        float* __restrict__ G) {
    __shared__ float Hs[16 * 64];
    const int row0 = blockIdx.x * 16;
    const int wave = threadIdx.x >> 5;
    const int lane = threadIdx.x & 31;
    const int half = lane >> 4;
    const int lr   = lane & 15;
    const int col0 = wave * 16;

    v8f c = {0.f, 0.f, 0.f, 0.f, 0.f, 0.f, 0.f, 0.f};
    const float* __restrict__ arow = E + (size_t)(row0 + lr) * 128;
#pragma unroll
    for (int kk = 0; kk < 128; kk += 4) {
        const int ke = kk + 2 * half;
        float2 af = *reinterpret_cast<const float2*>(arow + ke);
        v2f a = {af.x, af.y};
        v2f b = {We[(size_t)ke * 64 + col0 + lr],
                 We[(size_t)(ke + 1) * 64 + col0 + lr]};
        c = wmma4_f32(a, b, c);
    }
    const float bias = be[col0 + lr];
#pragma unroll
    for (int v = 0; v < 8; ++v)
        Hs[(v + 8 * half) * 64 + col0 + lr] = fmaxf(c[v] + bias, 0.f);
    __syncthreads();

    v8f g = {0.f, 0.f, 0.f, 0.f, 0.f, 0.f, 0.f, 0.f};
#pragma unroll
    for (int kk = 0; kk < 64; kk += 4) {
        const int ke = kk + 2 * half;
        v2f a = {Hs[lr * 64 + ke], Hs[lr * 64 + ke + 1]};
        v2f b = {W2[(size_t)ke * 64 + col0 + lr],
                 W2[(size_t)(ke + 1) * 64 + col0 + lr]};
        g = wmma4_f32(a, b, g);
    }
#pragma unroll
    for (int v = 0; v < 8; ++v)
        G[(size_t)(row0 + v + 8 * half) * 64 + col0 + lr] = g[v];
}

// ---------------------------------------------------------------------------
// Kernel 2: SheafBuilder pair MLP.
//   For pair q: h = relu([E[i]|E[j]] @ M1w + M1b) (256->64), p = h@M2w + M2b (64->2)
// Block = 128 threads / 16 pairs. The K=256 reduction is split into the E[i]
// half (M1w rows 0..127) and the E[j] half (rows 128..255), both WMMA.
// ---------------------------------------------------------------------------
__global__ void k_pairs(const float* __restrict__ E,  const float* __restrict__ M1w,
                        const float* __restrict__ M1b, const float* __restrict__ M2w,
                        const float* __restrict__ M2b,
                        const int* __restrict__ adj_i, const int* __restrict__ adj_j,
                        float* __restrict__ P) {
    __shared__ float Hs[16 * 64];
    const int q0   = blockIdx.x * 16;
    const int wave = threadIdx.x >> 5;
    const int lane = threadIdx.x & 31;
    const int half = lane >> 4;
    const int lr   = lane & 15;
    const int col0 = wave * 16;

    const int q = q0 + lr;
    const float* __restrict__ rowA = E + (size_t)adj_i[q] * 128;
    const float* __restrict__ rowB = E + (size_t)adj_j[q] * 128;

    v8f c = {0.f, 0.f, 0.f, 0.f, 0.f, 0.f, 0.f, 0.f};
#pragma unroll
    for (int kk = 0; kk < 128; kk += 4) {
        const int ke = kk + 2 * half;
        float2 af = *reinterpret_cast<const float2*>(rowA + ke);
        v2f a = {af.x, af.y};
        v2f b = {M1w[(size_t)ke * 64 + col0 + lr],
                 M1w[(size_t)(ke + 1) * 64 + col0 + lr]};
        c = wmma4_f32(a, b, c);
    }
#pragma unroll
    for (int kk = 0; kk < 128; kk += 4) {
        const int ke = kk + 2 * half;
        float2 af = *reinterpret_cast<const float2*>(rowB + ke);
        v2f a = {af.x, af.y};
        v2f b = {M1w[(size_t)(128 + ke) * 64 + col0 + lr],
                 M1w[(size_t)(128 + ke + 1) * 64 + col0 + lr]};
        c = wmma4_f32(a, b, c);
    }
    const float bias = M1b[col0 + lr];
#pragma unroll
    for (int v = 0; v < 8; ++v)
        Hs[(v + 8 * half) * 64 + col0 + lr] = fmaxf(c[v] + bias, 0.f);
    __syncthreads();

    // p = h @ M2w + M2b : 16 pairs x 2 outputs, one lane each.
    if (threadIdx.x < 32) {
        const int m  = threadIdx.x >> 1;
        const int aa = threadIdx.x & 1;
        float acc = M2b[aa];
#pragma unroll 8
        for (int cc = 0; cc < 64; ++cc) acc += Hs[m * 64 + cc] * M2w[cc * 2 + aa];
        P[(size_t)(q0 + m) * 2 + aa] = acc;
    }
}

// ---------------------------------------------------------------------------
// Kernel 3: s[i,a] = sum_t p[(i*8+t),a]^2 ; dinv = rsqrt(max(s+1e-5, 1e-6))
// ---------------------------------------------------------------------------
__global__ void k_seg(const float* __restrict__ P, float* __restrict__ S,
                      float* __restrict__ Dinv) {
    const int tid = blockIdx.x * blockDim.x + threadIdx.x;   // 0..8191 = i*2+a
    const int i = tid >> 1, a = tid & 1;
    float s = 0.f;
#pragma unroll
    for (int t = 0; t < 8; ++t) {
        float v = P[(size_t)(i * 8 + t) * 2 + a];
        s += v * v;
    }
    S[tid]    = s;
    Dinv[tid] = rsqrtf(fmaxf(s + 1e-5f, 1e-6f));
}

// ---------------------------------------------------------------------------
// Kernel 4: Y = elu((I - Delta) Xt) applied as a 9-point stencil over G rows,
// immediately column-reduced (deterministically, via LDS) into per-block
// partial sums. Xt[i*2+a,:] = w1s[a]*G[i,:] with w1s[a] = W1[0,a]+W1[1,a].
//   alpha_r = w1s[a]*(1 - dinv_r^2 * s_r)
//   beta_{r,t} = w1s[a]*dinv_r * p[rev(q),a]*p[q,a] * dinv[j*2+a]
// Block = 256 threads -> 4 rows x 64 cols.
// ---------------------------------------------------------------------------
__global__ void k_y(const float* __restrict__ G, const float* __restrict__ P,
                    const float* __restrict__ S, const float* __restrict__ Dinv,
                    const float* __restrict__ W1,
                    const int* __restrict__ adj_j, const int* __restrict__ rev_idx,
                    float* __restrict__ partial) {
    __shared__ float coef[4][8];
    __shared__ int   nidx[4][8];
    __shared__ float alpha[4];
    __shared__ int   irow[4];
    __shared__ float ybuf[4][64];

    const int r0  = blockIdx.x * 4;
    const int tid = threadIdx.x;

    if (tid < 32) {
        const int tr = tid >> 3, t = tid & 7;
        const int r = r0 + tr, i = r >> 1, a = r & 1;
        const int q = i * 8 + t;
        const int j = adj_j[q];
        const float w1s = W1[a] + W1[2 + a];
        const float dr  = Dinv[r];
        coef[tr][t] = w1s * dr * P[(size_t)rev_idx[q] * 2 + a] *
                      P[(size_t)q * 2 + a] * Dinv[j * 2 + a];
        nidx[tr][t] = j;
        if (t == 0) {
            alpha[tr] = w1s * (1.f - dr * dr * S[r]);
            irow[tr]  = i;
        }
    }
    __syncthreads();

    const int tr = tid >> 6, c = tid & 63;
    float acc = alpha[tr] * G[(size_t)irow[tr] * 64 + c];
#pragma unroll
    for (int t = 0; t < 8; ++t)
        acc += coef[tr][t] * G[(size_t)nidx[tr][t] * 64 + c];
    ybuf[tr][c] = (acc > 0.f) ? acc : expm1f(acc);   // elu
    __syncthreads();

    if (tid < 64)
        partial[(size_t)blockIdx.x * 64 + tid] =
            ybuf[0][tid] + ybuf[1][tid] + ybuf[2][tid] + ybuf[3][tid];
}

// ---------------------------------------------------------------------------
// Kernel 5: pooled mean + classifier. One block of 128 threads.
// ---------------------------------------------------------------------------
__global__ void k_cls(const float* __restrict__ partial, int nblocks,
                      const float* __restrict__ Wc1, const float* __restrict__ bc1,
                      const float* __restrict__ Wc2, const float* __restrict__ bc2,
                      float* __restrict__ out) {
    __shared__ float pooled[64];
    __shared__ float z[128];
    const int tid = threadIdx.x;

    if (tid < 64) {
        float s = 0.f;
        for (int b = 0; b < nblocks; ++b) s += partial[(size_t)b * 64 + tid];
        pooled[tid] = s * (1.f / 8192.f);
    }
    __syncthreads();
    {
        float acc = bc1[tid];
#pragma unroll 8
        for (int cc = 0; cc < 64; ++cc) acc += pooled[cc] * Wc1[(size_t)cc * 128 + tid];
        z[tid] = fmaxf(acc, 0.f);
    }
    __syncthreads();
    if (tid < 8) {
        float acc = bc2[tid];
#pragma unroll 8
        for (int h = 0; h < 128; ++h) acc += z[h] * Wc2[(size_t)h * 8 + tid];
        out[tid] = acc;
    }
}

// ---------------------------------------------------------------------------
// Host-side launcher
// ---------------------------------------------------------------------------
extern "C" void kernel_launch(void* const* d_in, const int* in_sizes, int n_in,
                              void* d_out, int out_size, void* d_ws, size_t ws_size,
                              hipStream_t stream) {
    (void)in_sizes; (void)n_in; (void)out_size; (void)ws_size;

    const float* E    = (const float*)d_in[0];   // (4096,128)
    const float* We   = (const float*)d_in[1];   // (128,64)
    const float* be   = (const float*)d_in[2];   // (64)
    const float* M1w  = (const float*)d_in[3];   // (256,64)
    const float* M1b  = (const float*)d_in[4];   // (64)
    const float* M2w  = (const float*)d_in[5];   // (64,2)
    const float* M2b  = (const float*)d_in[6];   // (2)
    const float* W1   = (const float*)d_in[7];   // (2,2)
    const float* W2   = (const float*)d_in[8];   // (64,64)
    const float* Wc1  = (const float*)d_in[9];   // (64,128)
    const float* bc1  = (const float*)d_in[10];  // (128)
    const float* Wc2  = (const float*)d_in[11];  // (128,8)
    const float* bc2  = (const float*)d_in[12];  // (8)
    const int* adj_i  = (const int*)d_in[13];    // (32768)
    const int* adj_j  = (const int*)d_in[14];    // (32768)
    const int* rev    = (const int*)d_in[15];    // (32768)
    float* out        = (float*)d_out;           // (8) f32 logits

    // Workspace layout (floats): ~1.86 MB total
    float* ws      = (float*)d_ws;
    float* Pp      = ws;                          // 32768*2   = 65536
    float* G       = Pp + 65536;                  // 4096*64   = 262144
    float* S       = G + 262144;                  // 8192
    float* Dinv    = S + 8192;                    // 8192
    float* partial = Dinv + 8192;                 // 2048*64   = 131072

    // 1) G = relu(E@We+be) @ W2      (WMMA f32, 256 blocks x 4 waves)
    hipLaunchKernelGGL(k_embed, dim3(4096 / 16), dim3(128), 0, stream,
                       E, We, be, W2, G);
    // 2) pair MLP -> p               (WMMA f32, 2048 blocks x 4 waves)
    hipLaunchKernelGGL(k_pairs, dim3(32768 / 16), dim3(128), 0, stream,
                       E, M1w, M1b, M2w, M2b, adj_i, adj_j, Pp);
    // 3) segment sums -> s, dinv
    hipLaunchKernelGGL(k_seg, dim3(8192 / 256), dim3(256), 0, stream,
                       Pp, S, Dinv);
    // 4) sheaf stencil + elu + partial column sums (2048 blocks)
    hipLaunchKernelGGL(k_y, dim3(8192 / 4), dim3(256), 0, stream,
                       G, Pp, S, Dinv, W1, adj_j, rev, partial);
    // 5) pool + classifier -> logits
    hipLaunchKernelGGL(k_cls, dim3(1), dim3(128), 0, stream,
                       partial, 8192 / 4, Wc1, bc1, Wc2, bc2, out);
}